// HierarchicalLayerNorm_38431367364877
// MI455X (gfx1250) — compile-verified
//
#include <hip/hip_runtime.h>
#include <cstdint>

// HierarchicalLayerNorm: B=8, S=4096, D=1024, fp32.
// Memory-bound: ~256MB traffic -> ~11us floor at 23.3 TB/s. No WMMA (0.17 GFLOP total).
// Design: one wave32 per token; x kept in registers across both LN passes;
// gamma/beta table (9 rows x 1024 x 2 = 72KB) staged in LDS via CDNA5 async DMA
// (GVS form: saddr = kernarg SGPR pair, vaddr = 32-bit byte offset; branch-free).

typedef float v4f __attribute__((ext_vector_type(4)));   // native vector: OK for nontemporal builtins

#define DDIM 1024
#define VEC (DDIM / 128)          // 16B chunks per lane (wave32) = 8
#define ROW4 (DDIM / 4)           // 16B chunks per table row = 256
#define NROWS 9                   // 8 splat rows + 1 base row
#define TBL4 (NROWS * ROW4)       // 16B chunks per table = 2304
#define NUM_SPLATS 20
#define LN_EPS 1e-6f

// Async-DMA one contiguous segment of n4 16B-chunks from global src into LDS
// at 16B-chunk offset dst4. n4 must be a multiple of blockDim (uniform trips).
__device__ __forceinline__ void stage_segment(uint32_t ldsBase, const float* src,
                                              int dst4, int n4) {
    const uint64_t sbase = (uint64_t)(uintptr_t)src;     // stays in SGPRs (uniform)
    for (int i = threadIdx.x; i < n4; i += 256) {
        const uint32_t loff = ldsBase + (uint32_t)(dst4 + i) * 16u;
        const uint32_t voff = (uint32_t)i * 16u;
        asm volatile("global_load_async_to_lds_b128 %0, %1, %2"
                     :: "v"(loff), "v"(voff), "s"(sbase) : "memory");
    }
}

__global__ __launch_bounds__(256) void hier_layernorm_gfx1250(
    const float* __restrict__ x,
    const int*   __restrict__ assign,
    const float* __restrict__ splat_w,
    const float* __restrict__ splat_b,
    const float* __restrict__ base_w,
    const float* __restrict__ base_b,
    float* __restrict__ out,
    int ntok)
{
    extern __shared__ v4f lds4[];                 // [2*TBL4]: gamma table then beta table
    const uint32_t ldsBase = (uint32_t)(uintptr_t)lds4;

    // ---- Stage tables as 4 contiguous segments (no divergence, no 64-bit VALU adds).
    // Layout: [splat_w rows 0..7 | base_w][splat_b rows 0..7 | base_b]
    // so "row 8" of each table IS the base row — read side needs no special case.
    stage_segment(ldsBase, splat_w, 0,               8 * ROW4);  // 2048 chunks
    stage_segment(ldsBase, base_w,  8 * ROW4,        ROW4);      //  256 chunks
    stage_segment(ldsBase, splat_b, TBL4,            8 * ROW4);
    stage_segment(ldsBase, base_b,  TBL4 + 8 * ROW4, ROW4);
    asm volatile("s_wait_asynccnt 0x0" ::: "memory");
    __syncthreads();

    const int wave = threadIdx.x >> 5;
    const int lane = threadIdx.x & 31;
    const int tok  = blockIdx.x * 8 + wave;
    if (tok >= ntok) return;

    // ---- Pass 1: stream the token row into registers (coalesced b128, non-temporal) ----
    const v4f* __restrict__ xrow = (const v4f*)(x + (size_t)tok * DDIM);
    v4f v[VEC];
    float s = 0.0f;
#pragma unroll
    for (int k = 0; k < VEC; ++k) {
        v[k] = __builtin_nontemporal_load(&xrow[lane + k * 32]);
        s += (v[k].x + v[k].y) + (v[k].z + v[k].w);
    }
    // wave32 butterfly reduction for the mean
#pragma unroll
    for (int off = 16; off > 0; off >>= 1) s += __shfl_xor(s, off, 32);
    const float mean = s * (1.0f / DDIM);

    // ---- Pass 2 (registers only): variance of (x - mean), matching reference numerics ----
    float q = 0.0f;
#pragma unroll
    for (int k = 0; k < VEC; ++k) {
        float dx;
        dx = v[k].x - mean; q = fmaf(dx, dx, q);
        dx = v[k].y - mean; q = fmaf(dx, dx, q);
        dx = v[k].z - mean; q = fmaf(dx, dx, q);
        dx = v[k].w - mean; q = fmaf(dx, dx, q);
    }
#pragma unroll
    for (int off = 16; off > 0; off >>= 1) q += __shfl_xor(q, off, 32);
    const float rstd = rsqrtf(q * (1.0f / DDIM) + LN_EPS);

    // ---- Per-token affine row selection: 0..19 -> min(id,7); 20 -> base (row 8) ----
    const int a   = assign[tok];
    const int row = (a >= NUM_SPLATS) ? 8 : ((a < 7) ? a : 7);
    const v4f* __restrict__ gW = lds4 + row * ROW4;
    const v4f* __restrict__ gB = lds4 + TBL4 + row * ROW4;
    v4f* __restrict__ orow = (v4f*)(out + (size_t)tok * DDIM);

#pragma unroll
    for (int k = 0; k < VEC; ++k) {
        const v4f g = gW[lane + k * 32];   // ds_load_b128, conflict-free
        const v4f b = gB[lane + k * 32];
        v4f r;
        r.x = fmaf((v[k].x - mean) * rstd, g.x, b.x);
        r.y = fmaf((v[k].y - mean) * rstd, g.y, b.y);
        r.z = fmaf((v[k].z - mean) * rstd, g.z, b.z);
        r.w = fmaf((v[k].w - mean) * rstd, g.w, b.w);
        __builtin_nontemporal_store(r, &orow[lane + k * 32]);
    }
}

extern "C" void kernel_launch(void* const* d_in, const int* in_sizes, int n_in,
                              void* d_out, int out_size, void* d_ws, size_t ws_size,
                              hipStream_t stream) {
    (void)n_in; (void)out_size; (void)d_ws; (void)ws_size;
    const float* x        = (const float*)d_in[0];
    const int*   assign   = (const int*)d_in[1];
    const float* splat_w  = (const float*)d_in[2];
    const float* splat_b  = (const float*)d_in[3];
    const float* base_w   = (const float*)d_in[4];
    const float* base_b   = (const float*)d_in[5];
    float*       out      = (float*)d_out;

    const int ntok   = in_sizes[1];              // B*S tokens
    const int blocks = (ntok + 7) / 8;           // 8 waves (tokens) per 256-thread block
    const size_t shmem = (size_t)2 * TBL4 * sizeof(v4f);  // 73728 bytes

    hier_layernorm_gfx1250<<<blocks, 256, shmem, stream>>>(
        x, assign, splat_w, splat_b, base_w, base_b, out, ntok);
}